// QuantumTime_55490977464503
// MI455X (gfx1250) — compile-verified
//
#include <hip/hip_runtime.h>
#include <math.h>
#include <stdint.h>

// ---- problem dims (match reference) ----
#define B_    128
#define T_    608
#define LB_   512
#define HZ_   96
#define INF_  7
#define LS_   16
#define RES_  256
#define LEAKF 0.3f
#define NR    257   // ridge system size (256 + bias)
#define NA    272   // padded row stride for A (multiple of 16)

typedef __attribute__((ext_vector_type(2))) float v2f;
typedef __attribute__((ext_vector_type(8))) float v8f;
typedef __attribute__((ext_vector_type(4))) unsigned int u32x4;
typedef __attribute__((ext_vector_type(8))) int i32x8;
typedef __attribute__((ext_vector_type(4))) int i32x4;

// therock/clang-23 lane ships this header and uses the 6-arg TDM builtin;
// ROCm 7.2 (clang-22) uses the 5-arg form.
#if defined(__has_include)
#if __has_include(<hip/amd_detail/amd_gfx1250_TDM.h>)
#define TDM_6ARG 1
#endif
#endif

// =====================================================================
// 1) coords -> h = coords @ feat_w + feat_b
// =====================================================================
__global__ void k_feat(const float* __restrict__ xt, const float* __restrict__ yt,
                       const float* __restrict__ fw, const float* __restrict__ fb,
                       float* __restrict__ h) {
  int n = blockIdx.x * blockDim.x + threadIdx.x;
  if (n >= B_ * T_) return;
  int b = n / T_, t = n - b * T_;
  float coord[8];
  coord[0] = (float)t / (float)(T_ - 1);
  const float* src = (t < LB_) ? (xt + ((size_t)b * LB_ + t) * INF_)
                               : (yt + ((size_t)b * HZ_ + (t - LB_)) * INF_);
#pragma unroll
  for (int k = 0; k < INF_; k++) coord[k + 1] = src[k];
  float* hr = h + (size_t)n * LS_;
#pragma unroll
  for (int j = 0; j < LS_; j++) {
    float acc = fb[j];
#pragma unroll
    for (int k = 0; k < 8; k++) acc = fmaf(coord[k], fw[k * LS_ + j], acc);
    hr[j] = acc;
  }
}

// =====================================================================
// 2) q = h @ qlin_w + qlin_b ;  c = relu(h @ clin_w + clin_b)
// =====================================================================
__global__ void k_qc(const float* __restrict__ h,
                     const float* __restrict__ qw, const float* __restrict__ qb,
                     const float* __restrict__ cw, const float* __restrict__ cb,
                     float* __restrict__ q, float* __restrict__ c) {
  int n = blockIdx.x * blockDim.x + threadIdx.x;
  if (n >= B_ * T_) return;
  float hr[LS_];
  const float* hp = h + (size_t)n * LS_;
#pragma unroll
  for (int k = 0; k < LS_; k++) hr[k] = hp[k];
#pragma unroll
  for (int j = 0; j < LS_; j++) {
    float qa = qb[j], ca = cb[j];
#pragma unroll
    for (int k = 0; k < LS_; k++) {
      qa = fmaf(hr[k], qw[k * LS_ + j], qa);
      ca = fmaf(hr[k], cw[k * LS_ + j], ca);
    }
    q[(size_t)n * LS_ + j] = qa;
    c[(size_t)n * LS_ + j] = fmaxf(ca, 0.0f);
  }
}

// =====================================================================
// 3) batchnorm over (B, LS_) per time-step t (in place on q)
// =====================================================================
__global__ void k_bn(float* __restrict__ q, const float* __restrict__ gamma,
                     const float* __restrict__ beta) {
  int t = blockIdx.x;
  int tid = threadIdx.x;
  __shared__ float red[256];
  float s = 0.f, s2 = 0.f;
  for (int m = tid; m < B_ * LS_; m += 256) {
    int b = m >> 4, j = m & 15;
    float v = q[((size_t)b * T_ + t) * LS_ + j];
    s += v; s2 += v * v;
  }
  red[tid] = s; __syncthreads();
  for (int o = 128; o > 0; o >>= 1) { if (tid < o) red[tid] += red[tid + o]; __syncthreads(); }
  float mean = red[0] * (1.0f / (B_ * LS_));
  __syncthreads();
  red[tid] = s2; __syncthreads();
  for (int o = 128; o > 0; o >>= 1) { if (tid < o) red[tid] += red[tid + o]; __syncthreads(); }
  float var = red[0] * (1.0f / (B_ * LS_)) - mean * mean;
  float inv = rsqrtf(var + 1e-5f);
  float g = gamma[t], be = beta[t];
  for (int m = tid; m < B_ * LS_; m += 256) {
    int b = m >> 4, j = m & 15;
    size_t idx = ((size_t)b * T_ + t) * LS_ + j;
    q[idx] = g * (q[idx] - mean) * inv + be;
  }
}

// =====================================================================
// 4) quantum layer: 4-wire state vector (16 complex amps) per sample
// =====================================================================
__device__ __forceinline__ void g_rx(float* re, float* im, int w, float th) {
  float c = cosf(0.5f * th), s = sinf(0.5f * th);
  int st = 1 << (3 - w);
#pragma unroll
  for (int i = 0; i < 16; i++) if (!(i & st)) {
    int j = i | st;
    float r0 = re[i], i0 = im[i], r1 = re[j], i1 = im[j];
    re[i] = c * r0 + s * i1;  im[i] = c * i0 - s * r1;
    re[j] = s * i0 + c * r1;  im[j] = -s * r0 + c * i1;
  }
}
__device__ __forceinline__ void g_ry(float* re, float* im, int w, float th) {
  float c = cosf(0.5f * th), s = sinf(0.5f * th);
  int st = 1 << (3 - w);
#pragma unroll
  for (int i = 0; i < 16; i++) if (!(i & st)) {
    int j = i | st;
    float r0 = re[i], i0 = im[i], r1 = re[j], i1 = im[j];
    re[i] = c * r0 - s * r1;  im[i] = c * i0 - s * i1;
    re[j] = s * r0 + c * r1;  im[j] = s * i0 + c * i1;
  }
}
__device__ __forceinline__ void g_rz(float* re, float* im, int w, float th) {
  float c = cosf(0.5f * th), s = sinf(0.5f * th);
  int st = 1 << (3 - w);
#pragma unroll
  for (int i = 0; i < 16; i++) {
    float r = re[i], q = im[i];
    if (i & st) { re[i] = c * r - s * q; im[i] = c * q + s * r; }     // * e^{+i t/2}
    else        { re[i] = c * r + s * q; im[i] = c * q - s * r; }     // * e^{-i t/2}
  }
}
__device__ __forceinline__ void g_h(float* re, float* im, int w) {
  const float f = 0.70710678118654752f;
  int st = 1 << (3 - w);
#pragma unroll
  for (int i = 0; i < 16; i++) if (!(i & st)) {
    int j = i | st;
    float r0 = re[i], i0 = im[i], r1 = re[j], i1 = im[j];
    re[i] = f * (r0 + r1); im[i] = f * (i0 + i1);
    re[j] = f * (r0 - r1); im[j] = f * (i0 - i1);
  }
}
__device__ __forceinline__ void g_sx(float* re, float* im, int w) {
  int st = 1 << (3 - w);
#pragma unroll
  for (int i = 0; i < 16; i++) if (!(i & st)) {
    int j = i | st;
    float r0 = re[i], i0 = im[i], r1 = re[j], i1 = im[j];
    re[i] = 0.5f * ((r0 - i0) + (r1 + i1));
    im[i] = 0.5f * ((r0 + i0) + (i1 - r1));
    re[j] = 0.5f * ((r0 + i0) + (r1 - i1));
    im[j] = 0.5f * ((i0 - r0) + (r1 + i1));
  }
}
__device__ __forceinline__ void g_cnot(float* re, float* im, int c, int t) {
  int cst = 1 << (3 - c), tst = 1 << (3 - t);
#pragma unroll
  for (int i = 0; i < 16; i++) if ((i & cst) && !(i & tst)) {
    int j = i | tst;
    float tr = re[i], ti = im[i];
    re[i] = re[j]; im[i] = im[j];
    re[j] = tr;    im[j] = ti;
  }
}
__device__ __forceinline__ void g_meas(const float* re, const float* im, float* out) {
#pragma unroll
  for (int w = 0; w < 4; w++) {
    int st = 1 << (3 - w);
    float s = 0.f;
#pragma unroll
    for (int i = 0; i < 16; i++) {
      float p = re[i] * re[i] + im[i] * im[i];
      s += (i & st) ? -p : p;
    }
    out[w] = s;
  }
}

__global__ void k_quantum(const float* __restrict__ q,
                          const float* __restrict__ rz1, const float* __restrict__ ry1,
                          const float* __restrict__ rz2, const float* __restrict__ rz3,
                          float* __restrict__ qout) {
  int n = blockIdx.x * blockDim.x + threadIdx.x;
  if (n >= B_ * T_) return;
  float x[LS_];
  const float* qp = q + (size_t)n * LS_;
#pragma unroll
  for (int k = 0; k < LS_; k++) x[k] = qp[k];
  float re[16], im[16];
#pragma unroll
  for (int i = 0; i < 16; i++) { re[i] = 0.f; im[i] = 0.f; }
  re[0] = 1.f;
#pragma unroll
  for (int k = 0; k < 2; k++) {
#pragma unroll
    for (int i = 0; i < 4; i++) g_rz(re, im, i, rz1[k * 4 + i]);
#pragma unroll
    for (int i = 0; i < 4; i++) g_ry(re, im, i, ry1[k * 4 + i]);
#pragma unroll
    for (int i = 0; i < 4; i++) g_rz(re, im, i, rz2[k * 4 + i]);
#pragma unroll
    for (int i = 0; i < 4; i++) g_cnot(re, im, i, (i + 1) & 3);
    if (k == 0) {
#pragma unroll
      for (int i = 0; i < 4; i++) {
        g_rx(re, im, i, x[4 * i + 0]);
        g_ry(re, im, i, x[4 * i + 1]);
        g_rz(re, im, i, x[4 * i + 2]);
        g_rx(re, im, i, x[4 * i + 3]);
      }
    }
  }
  float m[4];
  float* op = qout + (size_t)n * LS_;
  g_meas(re, im, m);
#pragma unroll
  for (int w = 0; w < 4; w++) op[w] = m[w];
#pragma unroll
  for (int i = 0; i < 4; i++) g_h(re, im, i);
  g_meas(re, im, m);
#pragma unroll
  for (int w = 0; w < 4; w++) op[4 + w] = m[w];
#pragma unroll
  for (int i = 0; i < 4; i++) { g_h(re, im, i); g_sx(re, im, i); }
  g_meas(re, im, m);
#pragma unroll
  for (int w = 0; w < 4; w++) op[8 + w] = m[w];
#pragma unroll
  for (int i = 0; i < 4; i++) g_rz(re, im, i, rz3[i]);
  g_meas(re, im, m);
#pragma unroll
  for (int w = 0; w < 4; w++) op[12 + w] = m[w];
}

// =====================================================================
// 5) h = concat(qout, c) @ lin_w + lin_b
// =====================================================================
__global__ void k_mix(const float* __restrict__ qo, const float* __restrict__ c,
                      const float* __restrict__ lw, const float* __restrict__ lb,
                      float* __restrict__ h) {
  int n = blockIdx.x * blockDim.x + threadIdx.x;
  if (n >= B_ * T_) return;
  float a[2 * LS_];
#pragma unroll
  for (int k = 0; k < LS_; k++) { a[k] = qo[(size_t)n * LS_ + k]; a[LS_ + k] = c[(size_t)n * LS_ + k]; }
#pragma unroll
  for (int j = 0; j < LS_; j++) {
    float acc = lb[j];
#pragma unroll
    for (int k = 0; k < 2 * LS_; k++) acc = fmaf(a[k], lw[k * LS_ + j], acc);
    h[(size_t)n * LS_ + j] = acc;
  }
}

// =====================================================================
// zero-fill helper
// =====================================================================
__global__ void k_fill(float* __restrict__ p, int n) {
  int i = blockIdx.x * blockDim.x + threadIdx.x;
  if (i < n) p[i] = 0.f;
}

// =====================================================================
// 6) reservoir step (WMMA f32 16x16x4):
//    Snew = (1-L)*S + L*sin(S @ W_res + X_j @ W_in)
//    block = 256 (8 waves), grid = (M/16, 2); wave handles one 16x16 D tile
//    S tile (16x256 f32) staged into LDS by the Tensor Data Mover with the
//    TDM pad feature reproducing the [16][260] bank-padded layout.
// =====================================================================
__global__ void __launch_bounds__(256)
k_res_step(const float* __restrict__ Sin, const float* __restrict__ Xbase,
           const float* __restrict__ Wres, const float* __restrict__ Win,
           float* __restrict__ Sout, float* __restrict__ RSout,
           int nw, int Tt, int tstride, int j) {
  __shared__ float sS[16][RES_ + 4];
  __shared__ float sX[16][LS_ + 4];
  const int row0 = blockIdx.x * 16;
  const int tid = threadIdx.x;

#if __has_builtin(__builtin_amdgcn_tensor_load_to_lds) && \
    __has_builtin(__builtin_amdgcn_s_wait_tensorcnt)
  if (tid < 32) {
    // D# group 0: count=1 (valid, user mode), lds_addr, 57-bit global addr, type=2
    unsigned long long ga =
        (unsigned long long)(uintptr_t)(Sin + (size_t)row0 * RES_);
    unsigned lds = (unsigned)(uintptr_t)&sS[0][0];   // low 32 bits = LDS byte offset
    u32x4 g0;
    g0[0] = 1u;
    g0[1] = lds;
    g0[2] = (unsigned)(ga & 0xffffffffull);
    g0[3] = (unsigned)((ga >> 32) & 0x1ffffffull) | (2u << 30);
    // D# group 1: data_size=4B (code 2); pad_enable, pad_interval=256 DW (code 7),
    // pad_amount=4 DW (code 3) -> LDS row stride 260 DWORDs; 2D tile 256x16.
    i32x8 g1;
    g1[0] = (int)((2u << 16) | (1u << 20) | (7u << 22) | (3u << 25));
    g1[1] = (int)(256u << 16);   // tensor_dim0 = 256 elements
    g1[2] = (int)(16u << 16);    // tensor_dim1 = 16 rows
    g1[3] = (int)(256u << 16);   // tile_dim0 = 256
    g1[4] = 16;                  // tile_dim1 = 16, tile_dim2 = 0
    g1[5] = 256;                 // tensor_dim0_stride = 256 (elements)
    g1[6] = (int)(256u << 16);   // tensor_dim1_stride[15:0] = 256
    g1[7] = 0;                   // tensor_dim1_stride[47:16] = 0
    i32x4 gz = {0, 0, 0, 0};
#ifdef TDM_6ARG
    i32x8 gz8 = {0, 0, 0, 0, 0, 0, 0, 0};
    __builtin_amdgcn_tensor_load_to_lds(g0, g1, gz, gz, gz8, 0);
#else
    __builtin_amdgcn_tensor_load_to_lds(g0, g1, gz, gz, 0);
#endif
    __builtin_amdgcn_s_wait_tensorcnt(0);
  }
#else
  for (int idx = tid; idx < 16 * RES_; idx += 256) {
    int m = idx >> 8, k = idx & 255;
    sS[m][k] = Sin[(size_t)(row0 + m) * RES_ + k];
  }
#endif

  // stage matching input rows (16x16, with zero padding outside [0,Tt))
  for (int idx = tid; idx < 16 * LS_; idx += 256) {
    int m = idx >> 4, k = idx & 15;
    int r = row0 + m, b = r / nw, i = r - b * nw;
    int tx = 4 * i + j - 2;
    sX[m][k] = (tx >= 0 && tx < Tt) ? Xbase[((size_t)b * tstride + tx) * LS_ + k] : 0.0f;
  }
  __syncthreads();

  const int wave = tid >> 5, lane = tid & 31;
  const int n0 = (blockIdx.y * 8 + wave) * 16;
  const int ml = lane & 15, half = lane >> 4;

  __builtin_prefetch(Wres + (size_t)n0, 0, 1);   // global_prefetch_b8

  v8f acc = {};
#pragma unroll 4
  for (int k0 = 0; k0 < RES_; k0 += 4) {
    int ka = k0 + 2 * half;
    v2f a, bf;
    a.x  = sS[ml][ka];       a.y  = sS[ml][ka + 1];
    bf.x = Wres[(size_t)ka * RES_ + n0 + ml];
    bf.y = Wres[(size_t)(ka + 1) * RES_ + n0 + ml];
    acc = __builtin_amdgcn_wmma_f32_16x16x4_f32(false, a, false, bf, (short)0, acc,
                                                false, false);
  }
#pragma unroll
  for (int k0 = 0; k0 < LS_; k0 += 4) {
    int ka = k0 + 2 * half;
    v2f a, bf;
    a.x  = sX[ml][ka];       a.y  = sX[ml][ka + 1];
    bf.x = Win[(size_t)ka * RES_ + n0 + ml];
    bf.y = Win[(size_t)(ka + 1) * RES_ + n0 + ml];
    acc = __builtin_amdgcn_wmma_f32_16x16x4_f32(false, a, false, bf, (short)0, acc,
                                                false, false);
  }
#pragma unroll
  for (int v = 0; v < 8; v++) {
    int m = v + 8 * half;
    int r = row0 + m;
    float sold = sS[m][n0 + ml];
    float snew = (1.0f - LEAKF) * sold + LEAKF * sinf(acc[v]);
    Sout[(size_t)r * RES_ + n0 + ml] = snew;
    if (j >= 2) {
      int b = r / nw, i = r - b * nw;
      RSout[((size_t)b * Tt + 4 * i + (j - 2)) * RES_ + n0 + ml] = snew;
    }
  }
}

// =====================================================================
// 7) Gram matrix A[b] = rs_b^T rs_b  (256x256 block), WMMA, one wave/tile
// =====================================================================
__global__ void __launch_bounds__(32)
k_gram(const float* __restrict__ rs, float* __restrict__ Apad) {
  const int b = blockIdx.z;
  const int i0 = blockIdx.x * 16, j0 = blockIdx.y * 16;
  const float* X = rs + (size_t)b * LB_ * RES_;
  const int lane = threadIdx.x & 31, ml = lane & 15, half = lane >> 4;
  v8f acc = {};
#pragma unroll 4
  for (int t0 = 0; t0 < LB_; t0 += 4) {
    int ka = t0 + 2 * half;
    v2f a, bf;
    a.x  = X[(size_t)ka * RES_ + i0 + ml];
    a.y  = X[(size_t)(ka + 1) * RES_ + i0 + ml];
    bf.x = X[(size_t)ka * RES_ + j0 + ml];
    bf.y = X[(size_t)(ka + 1) * RES_ + j0 + ml];
    acc = __builtin_amdgcn_wmma_f32_16x16x4_f32(false, a, false, bf, (short)0, acc,
                                                false, false);
  }
  float* A = Apad + (size_t)b * NA * NA;
#pragma unroll
  for (int v = 0; v < 8; v++)
    A[(size_t)(i0 + v + 8 * half) * NA + j0 + ml] = acc[v];
}

// border row/col (ones column), diagonal regularization
__global__ void k_ridge_fix(const float* __restrict__ rs, const float* __restrict__ lam,
                            float* __restrict__ Apad) {
  int b = blockIdx.x, tid = threadIdx.x;
  float* A = Apad + (size_t)b * NA * NA;
  const float* X = rs + (size_t)b * LB_ * RES_;
  float reg = log1pf(expf(lam[0]));   // softplus
  if (tid < 256) {
    float s = 0.f;
    for (int t = 0; t < LB_; t++) s += X[(size_t)t * RES_ + tid];
    A[(size_t)256 * NA + tid] = s;
    A[(size_t)tid * NA + 256] = s;
    A[(size_t)tid * NA + tid] += reg;
  } else if (tid == 256) {
    A[(size_t)256 * NA + 256] = (float)LB_ + reg;
  }
}

// Bm[b][d][o] = X^T y  (d==256 row is the ones column)
__global__ void k_bm(const float* __restrict__ rs, const float* __restrict__ x,
                     float* __restrict__ Bm) {
  int idx = blockIdx.x * blockDim.x + threadIdx.x;
  if (idx >= B_ * NR * INF_) return;
  int o = idx % INF_;
  int rest = idx / INF_;
  int d = rest % NR;
  int b = rest / NR;
  const float* X = rs + (size_t)b * LB_ * RES_;
  const float* xb = x + (size_t)b * LB_ * INF_;
  float s = 0.f;
  if (d < 256) {
    for (int t = 0; t < LB_; t++) s = fmaf(X[(size_t)t * RES_ + d], xb[t * INF_ + o], s);
  } else {
    for (int t = 0; t < LB_; t++) s += xb[t * INF_ + o];
  }
  Bm[(size_t)b * NA * 8 + d * 8 + o] = s;
}

// =====================================================================
// 8) per-batch SPD solve (Gaussian elim, no pivot) : one block per batch
//    solution overwrites Bm
// =====================================================================
__global__ void __launch_bounds__(256)
k_solve(float* __restrict__ Apad, float* __restrict__ Bm) {
  int b = blockIdx.x, tid = threadIdx.x;
  float* A = Apad + (size_t)b * NA * NA;
  float* R = Bm + (size_t)b * NA * 8;
  __shared__ float pinv;
  for (int p = 0; p < NR; p++) {
    if (tid == 0) pinv = 1.0f / A[(size_t)p * NA + p];
    __syncthreads();
    for (int r = p + 1 + tid; r < NR; r += 256) {
      float f = A[(size_t)r * NA + p] * pinv;
      for (int c2 = p + 1; c2 < NR; c2++)
        A[(size_t)r * NA + c2] -= f * A[(size_t)p * NA + c2];
#pragma unroll
      for (int o = 0; o < INF_; o++) R[r * 8 + o] -= f * R[p * 8 + o];
      A[(size_t)r * NA + p] = 0.f;
    }
    __syncthreads();
  }
  if (tid < INF_) {
    for (int p = NR - 1; p >= 0; p--) {
      float s = R[p * 8 + tid];
      for (int c2 = p + 1; c2 < NR; c2++)
        s -= A[(size_t)p * NA + c2] * R[c2 * 8 + tid];
      R[p * 8 + tid] = s / A[(size_t)p * NA + p];
    }
  }
}

// =====================================================================
// 9) out[b,t,o] = sum_d W[b,d,o] * hres[b,t,d] + bias[b,o]
// =====================================================================
__global__ void k_final(const float* __restrict__ Bm, const float* __restrict__ hres,
                        float* __restrict__ out) {
  int idx = blockIdx.x * blockDim.x + threadIdx.x;
  if (idx >= B_ * HZ_ * INF_) return;
  int o = idx % INF_;
  int t = (idx / INF_) % HZ_;
  int b = idx / (INF_ * HZ_);
  const float* R = Bm + (size_t)b * NA * 8;
  const float* hr = hres + ((size_t)b * HZ_ + t) * RES_;
  float s = R[256 * 8 + o];
  for (int d = 0; d < 256; d++) s = fmaf(R[d * 8 + o], hr[d], s);
  out[idx] = s;
}

// =====================================================================
// host orchestration
// =====================================================================
extern "C" void kernel_launch(void* const* d_in, const int* in_sizes, int n_in,
                              void* d_out, int out_size, void* d_ws, size_t ws_size,
                              hipStream_t stream) {
  const float* x      = (const float*)d_in[0];
  const float* x_time = (const float*)d_in[1];
  const float* y_time = (const float*)d_in[2];
  const float* feat_w = (const float*)d_in[3];
  const float* feat_b = (const float*)d_in[4];
  const float* qlin_w = (const float*)d_in[5];
  const float* qlin_b = (const float*)d_in[6];
  const float* clin_w = (const float*)d_in[7];
  const float* clin_b = (const float*)d_in[8];
  const float* lin_w  = (const float*)d_in[9];
  const float* lin_b  = (const float*)d_in[10];
  const float* bn_g   = (const float*)d_in[11];
  const float* bn_b   = (const float*)d_in[12];
  const float* rz1    = (const float*)d_in[13];
  const float* ry1    = (const float*)d_in[14];
  const float* rz2    = (const float*)d_in[15];
  const float* rz3    = (const float*)d_in[16];
  const float* W_res  = (const float*)d_in[17];
  const float* W_in   = (const float*)d_in[18];
  const float* lam    = (const float*)d_in[19];

  float* ws = (float*)d_ws;
  const size_t sBT = (size_t)B_ * T_ * LS_;
  float* h    = ws;
  float* q    = h + sBT;
  float* qo   = q + sBT;
  float* cb   = qo + sBT;
  const size_t sS = (size_t)B_ * (LB_ / 4) * RES_;   // largest S buffer (LB pass)
  float* S0   = cb + sBT;
  float* S1   = S0 + sS;
  float* rs   = S1 + sS;
  float* hres = rs + (size_t)B_ * LB_ * RES_;
  float* Apad = hres + (size_t)B_ * HZ_ * RES_;
  float* Bm   = Apad + (size_t)B_ * NA * NA;

  const int NBT = (B_ * T_ + 255) / 256;

  // features
  k_feat<<<NBT, 256, 0, stream>>>(x_time, y_time, feat_w, feat_b, h);

  // two layers
  for (int l = 0; l < 2; l++) {
    k_qc<<<NBT, 256, 0, stream>>>(h, qlin_w + l * LS_ * LS_, qlin_b + l * LS_,
                                  clin_w + l * LS_ * LS_, clin_b + l * LS_, q, cb);
    k_bn<<<T_, 256, 0, stream>>>(q, bn_g + l * T_, bn_b + l * T_);
    k_quantum<<<(B_ * T_ + 127) / 128, 128, 0, stream>>>(q, rz1 + l * 8, ry1 + l * 8,
                                                         rz2 + l * 8, rz3 + l * 4, qo);
    k_mix<<<NBT, 256, 0, stream>>>(qo, cb, lin_w + l * 2 * LS_ * LS_, lin_b + l * LS_, h);
  }

  // reservoir over lookback window (nw = 128, M = 16384)
  {
    int nw = LB_ / 4, M = B_ * nw;
    k_fill<<<(int)((M * (size_t)RES_ + 255) / 256), 256, 0, stream>>>(S0, M * RES_);
    float *si = S0, *so = S1;
    for (int j = 0; j < 6; j++) {
      dim3 g(M / 16, 2);
      k_res_step<<<g, 256, 0, stream>>>(si, h, W_res, W_in, so, rs, nw, LB_, T_, j);
      float* tmp = si; si = so; so = tmp;
    }
  }

  // ridge: A = X^T X + reg I ; Bm = X^T y ; solve
  k_gram<<<dim3(16, 16, B_), 32, 0, stream>>>(rs, Apad);
  k_ridge_fix<<<B_, 257, 0, stream>>>(rs, lam, Apad);
  k_bm<<<(B_ * NR * INF_ + 255) / 256, 256, 0, stream>>>(rs, x, Bm);
  k_solve<<<B_, 256, 0, stream>>>(Apad, Bm);

  // reservoir over horizon window (nw = 24, M = 3072)
  {
    int nw = HZ_ / 4, M = B_ * nw;
    k_fill<<<(int)((M * (size_t)RES_ + 255) / 256), 256, 0, stream>>>(S0, M * RES_);
    float *si = S0, *so = S1;
    for (int j = 0; j < 6; j++) {
      dim3 g(M / 16, 2);
      k_res_step<<<g, 256, 0, stream>>>(si, h + (size_t)LB_ * LS_, W_res, W_in, so, hres,
                                        nw, HZ_, T_, j);
      float* tmp = si; si = so; so = tmp;
    }
  }

  // readout
  k_final<<<(B_ * HZ_ * INF_ + 255) / 256, 256, 0, stream>>>(Bm, hres, (float*)d_out);
}